// LayerNorm_91010357002261
// MI455X (gfx1250) — compile-verified
//
#include <hip/hip_runtime.h>

// Equivariant irrep LayerNorm, IRREPS=[(256,0),(128,1),(64,2)], DIM_TOTAL=960.
// Memory-bound: 3-phase plan (reduce -> tiny table build -> fused apply).
// Wave reductions use V_WMMA_F32_16X16X4_F32 (exact fp32 matrix reduction).

#define ROWS      16384
#define COLS      960
#define COLS4     240          // float4 columns per row
#define BATCHES   4
#define BPB       480          // blocks per batch; 480*256 stride divisible by 240
#define TPB       256
#define EPSV      1e-5

typedef __attribute__((ext_vector_type(2))) float v2f;
typedef __attribute__((ext_vector_type(4))) float v4f;
typedef __attribute__((ext_vector_type(8))) float v8f;

// Exact fp32 sum of `acc` across all 32 lanes using the matrix unit.
// A layout (16x4 f32): lanes 0-15 hold K=0,1 in VGPR0/1; lanes 16-31 hold K=2,3.
// With A=(acc,0) and B=ones: D[m][n] = acc[m] + acc[m+16] for every n.
// Each lane sums its 8 D VGPRs (rows 0-7 on lanes 0-15, rows 8-15 on lanes 16-31),
// then one shfl_xor(16) completes the 32-lane sum.
__device__ __forceinline__ float wave_sum_wmma(float acc) {
    v2f a; a.x = acc;  a.y = 0.0f;
    v2f b; b.x = 1.0f; b.y = 1.0f;   // all-ones B matrix (layout independent)
    v8f c = {};
    c = __builtin_amdgcn_wmma_f32_16x16x4_f32(false, a, false, b,
                                              (short)0, c, false, false);
    float t = c[0] + c[1] + c[2] + c[3] + c[4] + c[5] + c[6] + c[7];
    t += __shfl_xor(t, 16, 32);
    return t;
}

// Pass 1: per-batch partial sums. q = [sum_l0, sumsq_l0, sumsq_l1, sumsq_l2].
__global__ void __launch_bounds__(TPB)
reduce_kernel(const v4f* __restrict__ x, double* __restrict__ sums) {
    const int b = blockIdx.y;
    const v4f* xb = x + (size_t)b * (ROWS * COLS4);
    const int tid0   = blockIdx.x * TPB + threadIdx.x;
    const int stride = BPB * TPB;                 // divisible by COLS4
    const int c4     = tid0 % COLS4;              // loop-invariant column

    float s0 = 0.f, q0 = 0.f, q1 = 0.f, q2 = 0.f;
    for (int idx = tid0; idx < ROWS * COLS4; idx += stride) {
        v4f v = xb[idx];
        float ss = v.x * v.x + v.y * v.y + v.z * v.z + v.w * v.w;
        if (c4 < 64)       { s0 += v.x + v.y + v.z + v.w; q0 += ss; }
        else if (c4 < 160) { q1 += ss; }
        else               { q2 += ss; }
    }

    // wave32 reductions on the matrix unit (EXEC is all-ones here: full blocks,
    // reconverged control flow).
    s0 = wave_sum_wmma(s0);
    q0 = wave_sum_wmma(q0);
    q1 = wave_sum_wmma(q1);
    q2 = wave_sum_wmma(q2);

    __shared__ float red[TPB / 32][4];
    const int wave = threadIdx.x >> 5, lane = threadIdx.x & 31;
    if (lane == 0) { red[wave][0] = s0; red[wave][1] = q0;
                     red[wave][2] = q1; red[wave][3] = q2; }
    __syncthreads();
    if (threadIdx.x < 4) {
        float t = 0.f;
        for (int w = 0; w < TPB / 32; ++w) t += red[w][threadIdx.x];
        atomicAdd(&sums[b * 4 + threadIdx.x], (double)t);  // global_atomic_add_f64
    }
}

// Pass 2: fold statistics + weight/bias into per-(batch,column) scale/shift.
__global__ void build_table(const float* __restrict__ weight,
                            const float* __restrict__ bias,
                            const double* __restrict__ sums,
                            float* __restrict__ table /* [4][2][COLS] */) {
    const int c = blockIdx.x * blockDim.x + threadIdx.x;
    const int b = blockIdx.y;
    if (c >= COLS) return;

    const double s0 = sums[b * 4 + 0], q0 = sums[b * 4 + 1];
    const double q1 = sums[b * 4 + 2], q2 = sums[b * 4 + 3];

    const double N0    = (double)ROWS * 256.0;
    const double mean0 = s0 / N0;
    const double var0  = q0 / N0 - mean0 * mean0;
    const double inv0  = 1.0 / sqrt(var0 + EPSV);
    const double inv1  = 1.0 / sqrt(q1 / ((double)ROWS * 128.0) + EPSV);
    const double inv2  = 1.0 / sqrt(q2 / ((double)ROWS * 64.0) + EPSV);

    float scale, shift;
    if (c < 256) {
        const float w = weight[c];
        scale = (float)inv0 * w;
        shift = (float)(-mean0 * inv0) * w + bias[c];
    } else if (c < 640) {
        const int ch = (c - 256) / 3;
        scale = (float)inv1 * weight[256 + ch];
        shift = 0.f;
    } else {
        const int ch = (c - 640) / 5;
        scale = (float)inv2 * weight[384 + ch];
        shift = 0.f;
    }
    table[(b * 2 + 0) * COLS + c] = scale;
    table[(b * 2 + 1) * COLS + c] = shift;
}

// Pass 3: out = x*scale + shift. Table staged in LDS; float4 traffic;
// non-temporal stores so output doesn't evict input lines from L2;
// reversed traversal to catch L2-resident tail of pass 1.
__global__ void __launch_bounds__(TPB)
apply_kernel(const v4f* __restrict__ x, const float* __restrict__ table,
             v4f* __restrict__ out) {
    const int b = blockIdx.y;
    __shared__ float s_scale[COLS];
    __shared__ float s_shift[COLS];
    for (int i = threadIdx.x; i < COLS; i += TPB) {
        s_scale[i] = table[(b * 2 + 0) * COLS + i];
        s_shift[i] = table[(b * 2 + 1) * COLS + i];
    }
    __syncthreads();

    const v4f* xb = x   + (size_t)b * (ROWS * COLS4);
    v4f*       ob = out + (size_t)b * (ROWS * COLS4);

    const int bx     = gridDim.x - 1 - blockIdx.x;       // reversed block order
    const int tid0   = bx * TPB + threadIdx.x;
    const int stride = BPB * TPB;                        // divisible by COLS4
    const int c4     = tid0 % COLS4;                     // loop-invariant column
    const int c      = 4 * c4;
    const float sx = s_scale[c],     hx = s_shift[c];
    const float sy = s_scale[c + 1], hy = s_shift[c + 1];
    const float sz = s_scale[c + 2], hz = s_shift[c + 2];
    const float sw = s_scale[c + 3], hw = s_shift[c + 3];

    const int iters = (ROWS * COLS4) / stride;           // exactly 32
    for (int it = iters - 1; it >= 0; --it) {            // high addresses first
        const int idx = tid0 + it * stride;
        v4f v = xb[idx];
        v4f r;
        r.x = v.x * sx + hx;
        r.y = v.y * sy + hy;
        r.z = v.z * sz + hz;
        r.w = v.w * sw + hw;
        __builtin_nontemporal_store(r, &ob[idx]);
    }
}

extern "C" void kernel_launch(void* const* d_in, const int* in_sizes, int n_in,
                              void* d_out, int out_size, void* d_ws, size_t ws_size,
                              hipStream_t stream) {
    const float* x      = (const float*)d_in[0];   // (4, 16384, 960) fp32
    const float* weight = (const float*)d_in[1];   // (448,)
    const float* bias   = (const float*)d_in[2];   // (256,)
    float*       out    = (float*)d_out;

    double* sums  = (double*)d_ws;                       // 16 doubles
    float*  table = (float*)((char*)d_ws + 256);         // 4*2*960 floats

    (void)hipMemsetAsync(d_ws, 0, 256, stream);          // zero the accumulators

    reduce_kernel<<<dim3(BPB, BATCHES), dim3(TPB), 0, stream>>>(
        (const v4f*)x, sums);

    build_table<<<dim3((COLS + 255) / 256, BATCHES), dim3(256), 0, stream>>>(
        weight, bias, sums, table);

    apply_kernel<<<dim3(BPB, BATCHES), dim3(TPB), 0, stream>>>(
        (const v4f*)x, table, (v4f*)out);
}